// PinesAlgorithmLayer_31361851195766
// MI455X (gfx1250) — compile-verified
//
#include <hip/hip_runtime.h>

typedef __attribute__((ext_vector_type(2))) float v2f;
typedef __attribute__((ext_vector_type(8))) float v8f;

#define MU_F   398600441800000.0f
#define AREF_F 6378136.3f

#define NDEG  64      // N (max degree l)
#define LDIM  67      // cBar/sBar row stride (N+3)
#define NCOL  65      // columns m = 0..64
#define NSTEP 2016    // packed inner-recursion entries: sum_{m=0}^{62} (63-m)

__device__ __forceinline__ int colStart(int m) { return (m * (127 - m)) >> 1; }

__global__ void __launch_bounds__(256)
pines_kernel(const float* __restrict__ inp,
             const float* __restrict__ cBar,
             const float* __restrict__ sBar,
             float* __restrict__ out, int B)
{
    // Batch-invariant tables, broadcast-read from LDS (~34 KB / block)
    __shared__ float4 hA[NCOL];     // {diag[m], sub[m+1]*diag[m+1], cBar[m][m], sBar[m][m]}
    __shared__ float2 hB[NCOL];     // {cBar[m+1][m], sBar[m+1][m]}
    __shared__ float4 steps[NSTEP]; // {n1[l][m], n2[l][m], cBar[l][m], sBar[l][m]}

    const int tid = threadIdx.x;

    // ---- cooperative setup (f64 like the reference's constant pipeline) ----
    for (int m = tid; m < NCOL; m += blockDim.x) {
        double dg = 1.0;  // diag[m] = cumprod f,  f[1]=sqrt3, f[l>=2]=sqrt((2l+1)/(2l))
        for (int j = 1; j <= m; ++j)
            dg *= (j == 1) ? sqrt(3.0) : sqrt((2.0 * j + 1.0) / (2.0 * j));
        float q = 0.f, c1 = 0.f, s1 = 0.f;
        if (m < NDEG) {
            const int l = m + 1;
            const double fn = (l == 1) ? sqrt(3.0) : sqrt((2.0 * l + 1.0) / (2.0 * l));
            const double sb = (l == 1) ? 1.0 : sqrt(2.0 * l); // sub[l]
            q  = (float)(sb * fn * dg);                       // sub[m+1]*diag[m+1]
            c1 = cBar[l * LDIM + m];
            s1 = sBar[l * LDIM + m];
        }
        hA[m] = make_float4((float)dg, q, cBar[m * LDIM + m], sBar[m * LDIM + m]);
        hB[m] = make_float2(c1, s1);
    }
    for (int i = tid; i < NSTEP; i += blockDim.x) {
        // invert colStart: largest m with m(127-m)/2 <= i
        int m = (int)((127.0f - sqrtf(16129.0f - 8.0f * (float)i)) * 0.5f);
        if (m < 0) m = 0;
        while (m < 62 && colStart(m + 1) <= i) ++m;
        while (m > 0 && colStart(m) > i) --m;
        const int l = m + 2 + (i - colStart(m));
        const double dl = (double)l, dm = (double)m;
        const double n1 = sqrt((2.0 * dl + 1.0) * (2.0 * dl - 1.0) /
                               ((dl - dm) * (dl + dm)));
        const double n2 = sqrt((dl + dm - 1.0) * (2.0 * dl + 1.0) * (dl - dm - 1.0) /
                               ((dl + dm) * (dl - dm) * (2.0 * dl - 3.0)));
        steps[i] = make_float4((float)n1, (float)n2,
                               cBar[l * LDIM + m], sBar[l * LDIM + m]);
    }
    __syncthreads();

    // ---- per-point column-major Pines recursion (one thread = one point) ----
    const int b = blockIdx.x * blockDim.x + tid;
    const bool active = (b < B);
    float r = 1.f, sx = 0.f, tx = 0.f, ux = 0.f;
    if (active) {
        r  = inp[b * 4 + 0];
        sx = inp[b * 4 + 1];
        tx = inp[b * 4 + 2];
        ux = inp[b * 4 + 3];
    }
    const float fac = MU_F / r;    // rhol[l] = fac * rho^l
    const float rho = AREF_F / r;

    float re = 1.f, im = 0.f;      // rE[m], iM[m] updated per column
    float rhom = 1.f;              // rho^m
    float pot  = 0.f;              // sum over l>=1 of rho^l * S_l (fac factored out)

    for (int m = 0; m < NCOL; ++m) {
        const float4 h  = hA[m];
        const float2 h2 = hB[m];
        if (m > 0) {
            const float re2 = sx * re - tx * im;  // trig recursion
            im = sx * im + tx * re;
            re = re2;
            rhom *= rho;
            // diagonal term l = m : A[m][m] = diag[m]
            pot = fmaf(rhom * h.x, fmaf(h.z, re, h.w * im), pot);
        }
        float p2 = h.x;            // A[m][m]
        float p1 = h.y * ux;       // A[m+1][m] = sub[m+1]*diag[m+1]*u  (0 at m=64)
        float rpow = rhom * rho;   // rho^(m+1)
        pot = fmaf(rpow * p1, fmaf(h2.x, re, h2.y * im), pot);
        const int base = colStart(m);
        const int cnt  = (m < 63) ? (63 - m) : 0;  // l = m+2 .. 64 (uniform trip count)
        for (int k = 0; k < cnt; ++k) {
            const float4 e = steps[base + k];      // broadcast ds_load_b128
            rpow *= rho;
            const float p = fmaf(ux, e.x * p1, -(e.y * p2));  // u*n1*p1 - n2*p2
            pot = fmaf(rpow * p, fmaf(e.z, re, e.w * im), pot);
            p2 = p1; p1 = p;
        }
    }

    // ---- WMMA epilogue: out = -(fac + fac*pot) via D = A x B, B rows = -1 ----
    const float fp = fac * pot;
    const unsigned lane = (unsigned)(tid & 31);
    const int waveBase  = b & ~31;           // wave-uniform (256-thread blocks)

    if (waveBase + 32 <= B) {                // wave-uniform branch: EXEC stays all-1s
        const float facHi = __shfl_xor(fac, 16, 32);
        const float fpHi  = __shfl_xor(fp, 16, 32);
        const bool  lo = lane < 16u;
        v2f a0, a1, bm;
        a0.x = lo ? fac   : 0.f;  a0.y = lo ? fp   : 0.f;  // rows = lanes 0..15
        a1.x = lo ? facHi : 0.f;  a1.y = lo ? fpHi : 0.f;  // rows = lanes 16..31
        bm.x = lo ? -1.f  : 0.f;  bm.y = bm.x;             // B[0][*] = B[1][*] = -1
        v8f c = {};
        v8f d0 = __builtin_amdgcn_wmma_f32_16x16x4_f32(
                     false, a0, false, bm, (short)0, c, false, false);
        v8f d1 = __builtin_amdgcn_wmma_f32_16x16x4_f32(
                     false, a1, false, bm, (short)0, c, false, false);
        // D row i lives in VGPR i: lanes 0-15 hold rows 0..7, lanes 16-31 rows 8..15
        if (lane == 0u || lane == 16u) {
            const int rowOff = (lane == 0u) ? 0 : 8;
            float4* o  = (float4*)(out + waveBase + rowOff);
            o[0] = make_float4(d0[0], d0[1], d0[2], d0[3]);
            o[1] = make_float4(d0[4], d0[5], d0[6], d0[7]);
            float4* o2 = (float4*)(out + waveBase + 16 + rowOff);
            o2[0] = make_float4(d1[0], d1[1], d1[2], d1[3]);
            o2[1] = make_float4(d1[4], d1[5], d1[6], d1[7]);
        }
    } else if (active) {
        out[b] = -(fac + fp);
    }
}

extern "C" void kernel_launch(void* const* d_in, const int* in_sizes, int n_in,
                              void* d_out, int out_size, void* d_ws, size_t ws_size,
                              hipStream_t stream) {
    const float* inp  = (const float*)d_in[0];   // (B,4): r, s, t, u
    const float* cBar = (const float*)d_in[1];   // (67,67)
    const float* sBar = (const float*)d_in[2];   // (67,67)
    float* out = (float*)d_out;                  // (B,1) f32
    const int B = in_sizes[0] / 4;
    const int blocks = (B + 255) / 256;
    pines_kernel<<<blocks, 256, 0, stream>>>(inp, cBar, sBar, out, B);
}